// GIN_87419764343121
// MI455X (gfx1250) — compile-verified
//
#include <hip/hip_runtime.h>

#define N_NODES   100000
#define F_IN      32
#define DIM       64
#define N_EDGES   1600000
#define N_GRAPHS  2000
#define N_CLASSES 10
#define N_LAYERS  4
#define BN_EPS    1e-5f

typedef float v2f __attribute__((ext_vector_type(2)));
typedef float v8f __attribute__((ext_vector_type(8)));

// ---------------------------------------------------------------------------
// WMMA GEMM:  Out[M x 64] = (A (+ A2)) [M x K] @ W[K x 64] + bias[64]
// One 128-thread block = 4 waves; block handles 16 rows, wave w handles
// columns [16w, 16w+16). K in steps of 4 via V_WMMA_F32_16X16X4_F32.
// VGPR layouts per CDNA5 ISA 7.12.2:
//   A 16x4 : lanes 0-15 -> {K, K+1}, lanes 16-31 -> {K+2, K+3}, M = lane%16
//   B 4x16 : same K split, N = lane%16
//   C/D    : acc[v] = element (row r0 + v + 8*(lane/16), col c0 + lane%16)
// ---------------------------------------------------------------------------
template<int K, bool FUSE>
__global__ __launch_bounds__(128)
void gemm_wmma_kernel(const float* __restrict__ A, const float* __restrict__ A2,
                      const float* __restrict__ W, const float* __restrict__ bias,
                      float* __restrict__ Out)
{
    const int lane = threadIdx.x & 31;
    const int wave = threadIdx.x >> 5;   // column tile 0..3
    const int half = lane >> 4;          // 0/1
    const int lm   = lane & 15;
    const int r0   = blockIdx.x * 16;
    const int c0   = wave * 16;

    const float bv = bias[c0 + lm];
    v8f acc;
#pragma unroll
    for (int v = 0; v < 8; ++v) acc[v] = bv;

    const int arow = (r0 + lm) * K;
#pragma unroll
    for (int k = 0; k < K; k += 4) {
        const int ka = k + 2 * half;
        v2f a, b;
        if (FUSE) {
            a.x = A[arow + ka]     + A2[arow + ka];
            a.y = A[arow + ka + 1] + A2[arow + ka + 1];
        } else {
            a.x = A[arow + ka];
            a.y = A[arow + ka + 1];
        }
        b.x = W[ka * DIM + c0 + lm];
        b.y = W[(ka + 1) * DIM + c0 + lm];
        acc = __builtin_amdgcn_wmma_f32_16x16x4_f32(false, a, false, b,
                                                    (short)0, acc, false, false);
    }
#pragma unroll
    for (int v = 0; v < 8; ++v)
        Out[(r0 + v + 8 * half) * DIM + c0 + lm] = acc[v];
}

// ---------------------------------------------------------------------------
// agg[dst[e]][f] += h[src[e]][f]   (one thread per (edge, feature))
// ---------------------------------------------------------------------------
__global__ __launch_bounds__(256)
void scatter_add_kernel(const float* __restrict__ h, const int* __restrict__ src,
                        const int* __restrict__ dst, float* __restrict__ agg)
{
    long tid = (long)blockIdx.x * blockDim.x + threadIdx.x;
    if (tid >= (long)N_EDGES * DIM) return;
    int f = (int)(tid & (DIM - 1));
    int e = (int)(tid >> 6);
    atomicAdd(&agg[dst[e] * DIM + f], h[src[e] * DIM + f]);
}

// ---------------------------------------------------------------------------
// pooled[batch[n]][f] += X[n][f]  (feat is a power of two; shift = log2(feat))
// ---------------------------------------------------------------------------
__global__ __launch_bounds__(256)
void pool_kernel(const float* __restrict__ X, const int* __restrict__ batch,
                 float* __restrict__ pooled, int feat, int shift)
{
    long tid = (long)blockIdx.x * blockDim.x + threadIdx.x;
    if (tid >= ((long)N_NODES << shift)) return;
    int f = (int)(tid & (feat - 1));
    int n = (int)(tid >> shift);
    atomicAdd(&pooled[batch[n] * feat + f], X[tid]);
}

// ---------------------------------------------------------------------------
// Column sums / sums-of-squares over N_NODES rows of X[N x 64].
// stats[0..63] = sum, stats[64..127] = sumsq (caller zeroes first).
// ---------------------------------------------------------------------------
__global__ __launch_bounds__(256)
void colstats_kernel(const float* __restrict__ X, float* __restrict__ stats)
{
    __shared__ float sh[512];
    const int tid  = threadIdx.x;
    const int f    = tid & 63;
    const int rgrp = tid >> 6;                 // 0..3
    float s = 0.f, q = 0.f;
    for (int r = blockIdx.x * 4 + rgrp; r < N_NODES; r += gridDim.x * 4) {
        float v = X[r * DIM + f];
        s += v; q += v * v;
    }
    sh[tid] = s; sh[256 + tid] = q;
    __syncthreads();
    if (tid < 64) {
        s = sh[tid] + sh[tid + 64] + sh[tid + 128] + sh[tid + 192];
        q = sh[256 + tid] + sh[320 + tid] + sh[384 + tid] + sh[448 + tid];
        atomicAdd(&stats[f], s);
        atomicAdd(&stats[64 + f], q);
    }
}

// ---------------------------------------------------------------------------
// Out = relu(gamma * (X - mu) * rsqrt(var + eps) + beta), biased variance
// ---------------------------------------------------------------------------
__global__ __launch_bounds__(256)
void bnrelu_kernel(const float* __restrict__ X, const float* __restrict__ stats,
                   const float* __restrict__ gamma, const float* __restrict__ beta,
                   float* __restrict__ Out)
{
    long tid = (long)blockIdx.x * blockDim.x + threadIdx.x;
    if (tid >= (long)N_NODES * DIM) return;
    int f = (int)(tid & (DIM - 1));
    const float inv_n = 1.0f / (float)N_NODES;
    float mu  = stats[f] * inv_n;
    float var = stats[64 + f] * inv_n - mu * mu;
    float v = (X[tid] - mu) * rsqrtf(var + BN_EPS);
    v = gamma[f] * v + beta[f];
    Out[tid] = v > 0.f ? v : 0.f;
}

// ---------------------------------------------------------------------------
// out[g][c] = fcb0[c] + sum_i fcb[i][c] + pooled0[g]@fcW0[:,c]
//           + sum_i pooledL[i][g]@fcW[i][:,c];  out tail gets y as float.
// ---------------------------------------------------------------------------
__global__ __launch_bounds__(256)
void head_kernel(const float* __restrict__ pooled0, const float* __restrict__ pooledL,
                 const float* __restrict__ fcW0, const float* __restrict__ fcb0,
                 const float* __restrict__ fcW, const float* __restrict__ fcb,
                 const int* __restrict__ y, float* __restrict__ out)
{
    int tid = blockIdx.x * blockDim.x + threadIdx.x;
    if (tid < N_GRAPHS * N_CLASSES) {
        int g = tid / N_CLASSES, c = tid % N_CLASSES;
        float acc = fcb0[c];
#pragma unroll
        for (int i = 0; i < N_LAYERS; ++i) acc += fcb[i * N_CLASSES + c];
        for (int k = 0; k < F_IN; ++k)
            acc += pooled0[g * F_IN + k] * fcW0[k * N_CLASSES + c];
        for (int i = 0; i < N_LAYERS; ++i) {
            const float* p = pooledL + (long)i * N_GRAPHS * DIM + (long)g * DIM;
            const float* w = fcW + (long)i * DIM * N_CLASSES;
            for (int k = 0; k < DIM; ++k)
                acc += p[k] * w[k * N_CLASSES + c];
        }
        out[tid] = acc;
    }
    if (tid < N_GRAPHS) out[N_GRAPHS * N_CLASSES + tid] = (float)y[tid];
}

// ---------------------------------------------------------------------------
extern "C" void kernel_launch(void* const* d_in, const int* in_sizes, int n_in,
                              void* d_out, int out_size, void* d_ws, size_t ws_size,
                              hipStream_t stream)
{
    const float* x     = (const float*)d_in[0];
    const int*   edge  = (const int*)d_in[1];
    const int*   src   = edge;
    const int*   dst   = edge + N_EDGES;
    const int*   batch = (const int*)d_in[2];
    const int*   y     = (const int*)d_in[3];
    const float* W_enc = (const float*)d_in[4];
    const float* b_enc = (const float*)d_in[5];
    const float* W1    = (const float*)d_in[6];
    const float* b1    = (const float*)d_in[7];
    const float* g1    = (const float*)d_in[8];
    const float* be1   = (const float*)d_in[9];
    const float* W2    = (const float*)d_in[10];
    const float* b2    = (const float*)d_in[11];
    const float* gout  = (const float*)d_in[12];
    const float* bout  = (const float*)d_in[13];
    const float* fcW0  = (const float*)d_in[14];
    const float* fcb0  = (const float*)d_in[15];
    const float* fcW   = (const float*)d_in[16];
    const float* fcb   = (const float*)d_in[17];

    const size_t HN = (size_t)N_NODES * DIM;       // 6.4M floats
    float* hA      = (float*)d_ws;                 // h (layer input/output)
    float* hB      = hA + HN;                      // agg, then z
    float* hC      = hB + HN;                      // m1, then m2
    float* stats   = hC + HN;                      // 128 floats
    float* pooled0 = stats + 128;                  // G*F_IN
    float* pooledL = pooled0 + N_GRAPHS * F_IN;    // L*G*DIM
    const size_t need = (3 * HN + 128 + (size_t)N_GRAPHS * F_IN +
                         (size_t)N_LAYERS * N_GRAPHS * DIM) * sizeof(float);
    if (ws_size < need) return;

    const int ROWT   = N_NODES / 16;                       // 6250 row tiles
    const int ELEMB  = (int)((HN + 255) / 256);            // elementwise blocks
    const int EDGEB  = (int)(((long)N_EDGES * DIM + 255) / 256);

    // zero pooled accumulators (contiguous region)
    hipMemsetAsync(pooled0, 0,
                   (N_GRAPHS * F_IN + N_LAYERS * N_GRAPHS * DIM) * sizeof(float),
                   stream);

    // encoder: h = x @ W_enc + b_enc
    gemm_wmma_kernel<F_IN, false><<<ROWT, 128, 0, stream>>>(x, nullptr, W_enc, b_enc, hA);

    for (int i = 0; i < N_LAYERS; ++i) {
        const float* W1i = W1 + (size_t)i * DIM * DIM;
        const float* W2i = W2 + (size_t)i * DIM * DIM;

        // agg = segment_sum(h[src], dst)
        hipMemsetAsync(hB, 0, HN * sizeof(float), stream);
        scatter_add_kernel<<<EDGEB, 256, 0, stream>>>(hA, src, dst, hB);

        // m1 = (h + agg) @ W1[i] + b1[i]
        gemm_wmma_kernel<DIM, true><<<ROWT, 128, 0, stream>>>(hA, hB, W1i, b1 + i * DIM, hC);

        // z = relu(bn(m1))
        hipMemsetAsync(stats, 0, 128 * sizeof(float), stream);
        colstats_kernel<<<1024, 256, 0, stream>>>(hC, stats);
        bnrelu_kernel<<<ELEMB, 256, 0, stream>>>(hC, stats, g1 + i * DIM, be1 + i * DIM, hB);

        // m2 = z @ W2[i] + b2[i]
        gemm_wmma_kernel<DIM, false><<<ROWT, 128, 0, stream>>>(hB, nullptr, W2i, b2 + i * DIM, hC);

        // h = relu(bn(m2))
        hipMemsetAsync(stats, 0, 128 * sizeof(float), stream);
        colstats_kernel<<<1024, 256, 0, stream>>>(hC, stats);
        bnrelu_kernel<<<ELEMB, 256, 0, stream>>>(hC, stats, gout + i * DIM, bout + i * DIM, hA);

        // pooledL[i] = segment_sum(h, batch_ids)
        pool_kernel<<<ELEMB, 256, 0, stream>>>(hA, batch, pooledL + (size_t)i * N_GRAPHS * DIM,
                                               DIM, 6);
    }

    // pooled0 = segment_sum(x, batch_ids)
    pool_kernel<<<(int)(((long)N_NODES * F_IN + 255) / 256), 256, 0, stream>>>(
        x, batch, pooled0, F_IN, 5);

    // readout + y passthrough
    head_kernel<<<(N_GRAPHS * N_CLASSES + 255) / 256, 256, 0, stream>>>(
        pooled0, pooledL, fcW0, fcb0, fcW, fcb, y, (float*)d_out);
}